// PS_roi_align_84619445666344
// MI455X (gfx1250) — compile-verified
//
#include <hip/hip_runtime.h>
#include <hip/hip_bf16.h>
#include <cmath>

// PS-ROI-Align (position-sensitive ROI align, bilinear sampling, max over 2x2
// sample grid). Gather/latency-bound. One block per ROI, dim3(7,7,5) = 245
// threads; each thread produces TWO outputs (ctop = z and z+5) so the
// sampling-coordinate / bilinear-weight math (which depends only on ph,pw) is
// computed once per pair. The second channel sits at a constant +4,014,080 B
// offset -> folds into the global_load signed 24-bit immediate, reusing the
// same address VGPRs. ROI row is blockIdx-uniform -> scalar s_load broadcast +
// scalar-float SALU. Feature map (16 MB) is fully L2-resident (192 MB).

#define SPATIAL_SCALE (1.0f / 16.0f)

constexpr int S_ = 7;                 // ROI_SIZE
constexpr int G_ = 2;                 // SAMPLING_RATIO
constexpr int P_ = 10;                // POOLED_DIM
constexpr int C_ = P_ * S_ * S_;      // 490 channels
constexpr int H_ = 64;
constexpr int W_ = 64;
constexpr int ZSPLIT = P_ / 2;                    // 5: threads cover ctop and ctop+5
constexpr int CH_OFF = ZSPLIT * S_ * S_ * H_ * W_; // element offset between the pair

__global__ __launch_bounds__(S_ * S_ * ZSPLIT) void ps_roi_align_kernel(
    const float* __restrict__ data,   // [B, C, H, W]
    const float* __restrict__ rois,   // [N, 5]
    float* __restrict__ out)          // [N, P, S, S]
{
    const int pw = threadIdx.x;       // 0..6
    const int ph = threadIdx.y;       // 0..6
    const int zz = threadIdx.z;       // 0..4  (ctop = zz and zz + 5)
    const int n  = blockIdx.x;        // ROI index (wave-uniform)

    // Uniform across the block -> scalar s_load broadcast + scalar-float SALU.
    const float* roi = rois + (size_t)n * 5;
    const int   b      = (int)roi[0];
    const float roi_sw = roi[1] * SPATIAL_SCALE;
    const float roi_sh = roi[2] * SPATIAL_SCALE;
    const float roi_ew = roi[3] * SPATIAL_SCALE;
    const float roi_eh = roi[4] * SPATIAL_SCALE;

    const float bin_w = fmaxf(roi_ew - roi_sw, 1.0f) / (float)S_;
    const float bin_h = fmaxf(roi_eh - roi_sh, 1.0f) / (float)S_;

    // First channel of the pair; second is c0 + 245 (element offset CH_OFF).
    const int c0 = (zz * S_ + ph) * S_ + pw;
    const float* __restrict__ base =
        data + ((size_t)b * C_ + (size_t)c0) * (size_t)(H_ * W_);

    float max0 = -INFINITY;
    float max1 = -INFINITY;

    #pragma unroll
    for (int iy = 0; iy < G_; ++iy) {
        float y  = roi_sh + ((float)ph + ((float)iy + 0.5f) / (float)G_) * bin_h;
        bool  vy = (y > -1.0f) && (y < (float)H_);
        float yc = fminf(fmaxf(y, 0.0f), (float)(H_ - 1));
        int   yl = (int)floorf(yc);
        int   yh = min(yl + 1, H_ - 1);
        float ly = yc - (float)yl;
        float hy = 1.0f - ly;
        int   row_l = yl * W_;
        int   row_h = yh * W_;

        #pragma unroll
        for (int ix = 0; ix < G_; ++ix) {
            float x  = roi_sw + ((float)pw + ((float)ix + 0.5f) / (float)G_) * bin_w;
            bool  valid = vy && (x > -1.0f) && (x < (float)W_);
            float xc = fminf(fmaxf(x, 0.0f), (float)(W_ - 1));
            int   xl = (int)floorf(xc);
            int   xh = min(xl + 1, W_ - 1);
            float lx = xc - (float)xl;
            float hx = 1.0f - lx;

            const int i00 = row_l + xl;
            const int i01 = row_l + xh;
            const int i10 = row_h + xl;
            const int i11 = row_h + xh;

            // Channel pair: same address VGPRs, second load uses the +CH_OFF
            // immediate offset. 8 L2-hit gathers per sample point.
            float a00 = base[i00];
            float a01 = base[i01];
            float a10 = base[i10];
            float a11 = base[i11];
            float b00 = base[i00 + CH_OFF];
            float b01 = base[i01 + CH_OFF];
            float b10 = base[i10 + CH_OFF];
            float b11 = base[i11 + CH_OFF];

            float w00 = hy * hx, w01 = hy * lx, w10 = ly * hx, w11 = ly * lx;

            float val0 = w00 * a00 + w01 * a01 + w10 * a10 + w11 * a11;
            float val1 = w00 * b00 + w01 * b01 + w10 * b10 + w11 * b11;
            val0 = valid ? val0 : 0.0f;
            val1 = valid ? val1 : 0.0f;
            max0 = fmaxf(max0, val0);
            max1 = fmaxf(max1, val1);
        }
    }

    // Two block-contiguous NT stores (ranges [0,245) and [245,490) of the ROI).
    float* dst = out + (size_t)n * C_ + (size_t)c0;
    __builtin_nontemporal_store(max0, dst);
    __builtin_nontemporal_store(max1, dst + ZSPLIT * S_ * S_);
}

extern "C" void kernel_launch(void* const* d_in, const int* in_sizes, int n_in,
                              void* d_out, int out_size, void* d_ws, size_t ws_size,
                              hipStream_t stream) {
    const float* bottom_data = (const float*)d_in[0];  // [2, 490, 64, 64] f32
    const float* bottom_rois = (const float*)d_in[1];  // [N, 5] f32
    float* out = (float*)d_out;                        // [N, 10, 7, 7] f32

    int N = in_sizes[1] / 5;                           // number of ROIs
    if (N < 1) N = 1;

    dim3 block(S_, S_, ZSPLIT);                        // 7 x 7 x 5 = 245 threads
    dim3 grid(N);

    ps_roi_align_kernel<<<grid, block, 0, stream>>>(
        bottom_data, bottom_rois, out);
}